// FisheyeProjectionNet_76312978915631
// MI455X (gfx1250) — compile-verified
//
#include <hip/hip_runtime.h>

// heatmap[b,j,y,x] = g1(y-cy)*g1(x-cx), g1(d)=exp(-0.08*d^2) for |d|<=5 else 0.
// (outer(g1,g1)/sum/max == outer(g1,g1) since the peak is already 1.)
//
// Kernel 1: 1344 fisheye projections -> clipped int centers in d_ws.
// Kernel 2: one wave per 16x16 output tile. Wave-uniform scalar branch
//           (readfirstlane -> s_cbranch, EXEC untouched) skips the ~98% of
//           tiles outside the 11x11 window (pure zero stores). Hit tiles
//           compute the rank-1 tile with V_WMMA_F32_16X16X4_F32 and store
//           the f32 C/D layout with 8 coalesced non-temporal b32 stores.

typedef float v2f __attribute__((ext_vector_type(2)));
typedef float v8f __attribute__((ext_vector_type(8)));

#define IMG 256
#define TILES_PER_IMG 256   // (256/16)^2
#define WAVES_PER_BLOCK 8   // 256 threads / wave32

// ---- Kernel 1: per-(b,j) projection (full-precision libm, runs once) ----
__global__ __launch_bounds__(256)
void fisheye_centers(const float* __restrict__ joint,
                     int2* __restrict__ ctr, int nimg)
{
    const int i = blockIdx.x * blockDim.x + threadIdx.x;
    if (i >= nimg) return;
    const float jx = joint[i * 3 + 0];
    const float jy = joint[i * 3 + 1];
    const float jz = joint[i * 3 + 2];
    const float theta = atan2f(sqrtf(jx * jx + jy * jy), jz);
    const float phi   = atan2f(jy, jx);
    const float r     = 128.0f * theta / 1.57079632679489662f; // RADIUS*theta/(pi/2)
    int cx = (int)rintf(128.0f + r * cosf(phi));  // rintf == round-half-even (jnp.round)
    int cy = (int)rintf(128.0f + r * sinf(phi));
    cx = cx < 0 ? 0 : (cx > IMG - 1 ? IMG - 1 : cx);
    cy = cy < 0 ? 0 : (cy > IMG - 1 ? IMG - 1 : cy);
    ctr[i] = make_int2(cx, cy);
}

// ---- Kernel 2: tile writer (store-bandwidth bound) ----
__global__ __launch_bounds__(256)
void fisheye_heatmap_wmma(const int2* __restrict__ ctr,
                          float* __restrict__ out)
{
    const int wave = (blockIdx.x << 3) + (threadIdx.x >> 5);
    const int img  = wave >> 8;          // 256 tiles per image
    const int tile = wave & 255;
    const int ty   = tile >> 4;
    const int tx   = tile & 15;

    const int  lane  = threadIdx.x & 31;
    const int  laneN = lane & 15;
    const bool lo    = lane < 16;

    const int2 c  = ctr[img];            // same address across wave -> 1 request
    const int  cx = c.x, cy = c.y;
    const int  y0 = ty << 4, x0 = tx << 4;

    v8f d = {};

    // Wave-uniform 11x11-window intersection test; force a scalar branch so
    // EXEC stays all-ones through the WMMA (ISA requirement).
    const int hit = (y0 <= cy + 5) & (y0 + 15 >= cy - 5) &
                    (x0 <= cx + 5) & (x0 + 15 >= cx - 5);
    if (__builtin_amdgcn_readfirstlane(hit)) {
        const int dy = y0 + laneN - cy;
        const int dx = x0 + laneN - cx;
        // branchless window + fast exp (v_exp_f32), keeps EXEC untouched
        const float gy = (dy * dy <= 25) ? __expf((float)(dy * dy) * -0.08f) : 0.0f;
        const float gx = (dx * dx <= 25) ? __expf((float)(dx * dx) * -0.08f) : 0.0f;

        // A (16x4 f32): K=0 column = gy for M=lane (lanes 0-15); rest zero.
        // B (4x16 f32): K=0 row    = gx for N=lane (lanes 0-15); rest zero.
        v2f a; a.x = lo ? gy : 0.0f; a.y = 0.0f;
        v2f b; b.x = lo ? gx : 0.0f; b.y = 0.0f;
        v8f cc = {};
        // D[m][n] = gy[m] * gx[n]
        d = __builtin_amdgcn_wmma_f32_16x16x4_f32(
            false, a, false, b, (short)0, cc, false, false);
    }

    // C/D layout: VGPR v, lane l -> (M = v + 8*(l>=16), N = l%16)
    float* outp = out + ((size_t)img << 16)            // img * 256*256
                      + (size_t)(y0 + (lo ? 0 : 8)) * IMG
                      + (size_t)(x0 + laneN);
#pragma unroll
    for (int v = 0; v < 8; ++v)
        __builtin_nontemporal_store(d[v], outp + (size_t)v * IMG);
}

extern "C" void kernel_launch(void* const* d_in, const int* in_sizes, int n_in,
                              void* d_out, int out_size, void* d_ws, size_t ws_size,
                              hipStream_t stream)
{
    const float* joint = (const float*)d_in[0];   // [64, 21, 3] f32
    float*       out   = (float*)d_out;           // [64, 21, 256, 256] f32
    int2*        ctr   = (int2*)d_ws;             // 1344 * 8 B = 10.75 KB scratch

    const int nimg = in_sizes[0] / 3;             // 64*21 = 1344

    fisheye_centers<<<(nimg + 255) / 256, 256, 0, stream>>>(joint, ctr, nimg);

    const int blocks = nimg * (TILES_PER_IMG / WAVES_PER_BLOCK);  // 43008
    fisheye_heatmap_wmma<<<blocks, 256, 0, stream>>>(ctr, out);
}